// nchwBRA_61383672594824
// MI455X (gfx1250) — compile-verified
//
#include <hip/hip_runtime.h>
#include <hip/hip_bf16.h>
#include <math.h>

// ---------------------------------------------------------------------------
// BiFormer Bi-level Routing Attention for gfx1250 (MI455X), fp32 via
// V_WMMA_F32_16X16X4_F32, with async-to-LDS (ASYNCcnt) staging of the
// gathered K/V tiles in the attention kernel. Wave32 everywhere.
// ---------------------------------------------------------------------------

typedef __attribute__((ext_vector_type(2))) float v2f;
typedef __attribute__((ext_vector_type(8))) float v8f;

#define DIM      512
#define HEADS    8
#define HEAD_DIM 64
#define NB       8          // batch
#define HH       56
#define WW       56
#define NPIX     (HH*WW)    // 3136
#define NREG     49         // 7x7 regions
#define RSZ      64         // 8x8 tokens per region
#define TOPK     4
#define NKEYS    (TOPK*RSZ) // 256
#define MROWS    (NB*NPIX)  // 25088
#define SCALEF   (0.044194173824159216f) // 512^-0.5

__device__ __forceinline__ v8f wmma_f32x4(v2f a, v2f b, v8f c) {
  // D = A(16x4 f32) * B(4x16 f32) + C(16x16 f32)
  return __builtin_amdgcn_wmma_f32_16x16x4_f32(false, a, false, b,
                                               (short)0, c, false, false);
}

// Async copy of one contiguous 4KB tile (16 tokens x 64 f32) into LDS.
// Each lane provides its own LDS byte address and global address; the
// instruction's INST_OFFSET is added to BOTH sides (CDNA5 ISA),
// so 8 issues x 32 lanes x 16B cover the 4KB block. Tracked by ASYNCcnt.
__device__ __forceinline__ void async_copy_4k(unsigned ldst, const float* gsrc) {
  asm volatile("global_load_async_to_lds_b128 %0, %1, off"             :: "v"(ldst), "v"(gsrc) : "memory");
  asm volatile("global_load_async_to_lds_b128 %0, %1, off offset:512"  :: "v"(ldst), "v"(gsrc) : "memory");
  asm volatile("global_load_async_to_lds_b128 %0, %1, off offset:1024" :: "v"(ldst), "v"(gsrc) : "memory");
  asm volatile("global_load_async_to_lds_b128 %0, %1, off offset:1536" :: "v"(ldst), "v"(gsrc) : "memory");
  asm volatile("global_load_async_to_lds_b128 %0, %1, off offset:2048" :: "v"(ldst), "v"(gsrc) : "memory");
  asm volatile("global_load_async_to_lds_b128 %0, %1, off offset:2560" :: "v"(ldst), "v"(gsrc) : "memory");
  asm volatile("global_load_async_to_lds_b128 %0, %1, off offset:3072" :: "v"(ldst), "v"(gsrc) : "memory");
  asm volatile("global_load_async_to_lds_b128 %0, %1, off offset:3584" :: "v"(ldst), "v"(gsrc) : "memory");
}
__device__ __forceinline__ void wait_async_le8() {
  asm volatile("s_wait_asynccnt 0x8" ::: "memory");
}
__device__ __forceinline__ void wait_async_0() {
  asm volatile("s_wait_asynccnt 0x0" ::: "memory");
}

// ---------------------------------------------------------------------------
// Kernel 1: QKV projection  Y[25088,1536] = X[25088,512] * W^T + b,
// scattered directly into seq-layout q/k/v [B,HEADS,NREG,RSZ,HEAD_DIM].
// 8 waves / block, one 16x16 output tile per wave, K stepped by 4.
// ---------------------------------------------------------------------------
__global__ __launch_bounds__(256) void qkv_gemm_kernel(
    const float* __restrict__ X, const float* __restrict__ W,
    const float* __restrict__ bias,
    float* __restrict__ qseq, float* __restrict__ kseq, float* __restrict__ vseq) {
  const int lane = threadIdx.x & 31;
  const int wid  = blockIdx.x * 8 + (threadIdx.x >> 5);
  const int NT   = (3 * DIM) / 16;         // 96
  const int mt   = wid / NT;
  const int nt   = wid % NT;

  const int lrow = lane & 15;
  const int kb   = (lane >> 4) << 1;       // 0 or 2 (K sub-offset)
  const int row  = mt * 16 + lrow;
  const int col  = nt * 16 + lrow;
  const float* __restrict__ arow = X + (long)row * DIM;
  const float* __restrict__ brow = W + (long)col * DIM;  // B[k,n] = W[n,k]

  v8f acc = {};
#pragma unroll 4
  for (int k0 = 0; k0 < DIM; k0 += 4) {
    v2f a; a.x = arow[k0 + kb];     a.y = arow[k0 + kb + 1];
    v2f b; b.x = brow[k0 + kb];     b.y = brow[k0 + kb + 1];
    acc = wmma_f32x4(a, b, acc);
  }

  // scatter the 16x16 tile into seq layout
  const float bn   = bias[col];
  const int   arr  = col / DIM;            // 0=q 1=k 2=v
  const int   cc   = col % DIM;
  const int   head = cc >> 6, d = cc & 63;
  float* __restrict__ dst = (arr == 0) ? qseq : ((arr == 1) ? kseq : vseq);
  const int mbase = mt * 16 + ((lane >> 4) << 3);
#pragma unroll
  for (int r = 0; r < 8; ++r) {
    int grow = mbase + r;
    int b_   = grow / NPIX;
    int pix  = grow % NPIX;
    int hh   = pix / WW, ww = pix % WW;
    int reg  = (hh >> 3) * 7 + (ww >> 3);
    int tok  = ((hh & 7) << 3) + (ww & 7);
    dst[((((long)(b_ * HEADS + head) * NREG + reg) * RSZ + tok) * HEAD_DIM) + d]
        = acc[r] + bn;
  }
}

// ---------------------------------------------------------------------------
// Kernel 2: region mean-pool of q and k -> q_r/k_r [B,NREG,DIM]
// ---------------------------------------------------------------------------
__global__ void pool_kernel(const float* __restrict__ qseq,
                            const float* __restrict__ kseq,
                            float* __restrict__ qr, float* __restrict__ kr) {
  int i = blockIdx.x * blockDim.x + threadIdx.x;   // over B*NREG*DIM
  if (i >= NB * NREG * DIM) return;
  int c = i % DIM;
  int r = (i / DIM) % NREG;
  int b = i / (DIM * NREG);
  int head = c >> 6, d = c & 63;
  long base = (((long)(b * HEADS + head) * NREG + r) * RSZ) * HEAD_DIM + d;
  float sq = 0.f, sk = 0.f;
#pragma unroll 8
  for (int t = 0; t < RSZ; ++t) {
    sq += qseq[base + (long)t * HEAD_DIM];
    sk += kseq[base + (long)t * HEAD_DIM];
  }
  qr[i] = sq * (1.f / 64.f);
  kr[i] = sk * (1.f / 64.f);
}

// ---------------------------------------------------------------------------
// Kernel 3: routing scores a_r[b,i,j] = qr[b,i,:] . kr[b,j,:], top-4 indices.
// One block per (b, i); 64 threads.
// ---------------------------------------------------------------------------
__global__ void route_kernel(const float* __restrict__ qr,
                             const float* __restrict__ kr,
                             int* __restrict__ idxout) {
  __shared__ float sc[64];
  const int b = blockIdx.x / NREG;
  const int i = blockIdx.x % NREG;
  const int j = threadIdx.x;
  if (j < NREG) {
    const float* qa = qr + ((long)b * NREG + i) * DIM;
    const float* kb = kr + ((long)b * NREG + j) * DIM;
    float dot = 0.f;
#pragma unroll 8
    for (int c = 0; c < DIM; ++c) dot += qa[c] * kb[c];
    sc[j] = dot;
  }
  __syncthreads();
  if (threadIdx.x == 0) {
    int* out = idxout + ((long)b * NREG + i) * TOPK;
#pragma unroll
    for (int kk = 0; kk < TOPK; ++kk) {
      int best = 0; float bv = sc[0];
      for (int jj = 1; jj < NREG; ++jj)
        if (sc[jj] > bv) { bv = sc[jj]; best = jj; }
      out[kk] = best;
      sc[best] = -INFINITY;
    }
  }
}

// ---------------------------------------------------------------------------
// Kernel 4: gathered attention. One wave per (b, head, region, 16-row block).
// Gathered K/V tiles (each a contiguous 4KB block: a 16-wide n-tile never
// crosses a region boundary) are double-buffer streamed into LDS with
// global_load_async_to_lds_b128 (ASYNCcnt), overlapping DMA with WMMA.
// S = (Q*scale) Kg^T (16x256), softmax in LDS, O = P Vg (16x64).
// ---------------------------------------------------------------------------
#define QSTR 68
#define SSTR 260
__global__ __launch_bounds__(32) void attn_kernel(
    const float* __restrict__ qseq, const float* __restrict__ kseq,
    const float* __restrict__ vseq, const int* __restrict__ idx,
    float* __restrict__ attnout) {
  __shared__ float sQ[16 * QSTR];
  __shared__ float sS[16 * SSTR];
  __shared__ float sT[2][16 * 64];          // double-buffered K/V tile stage

  const int id   = blockIdx.x;
  const int mblk = id & 3;
  const int reg  = (id >> 2) % NREG;
  const int head = ((id >> 2) / NREG) % HEADS;
  const int b    = id / (4 * NREG * HEADS);
  const int lane = threadIdx.x;
  const int lrow = lane & 15;
  const int kb2  = (lane >> 4) << 1;

  // stage + scale the 16x64 Q block (contiguous 1024 floats)
  const float* __restrict__ qb =
      qseq + (((long)(b * HEADS + head) * NREG + reg) * RSZ + mblk * 16) * HEAD_DIM;
  for (int e = lane; e < 16 * 64; e += 32)
    sQ[(e >> 6) * QSTR + (e & 63)] = qb[e] * SCALEF;

  const int* __restrict__ idx4 = idx + ((long)b * NREG + reg) * TOPK;
  const int i0 = idx4[0], i1 = idx4[1], i2 = idx4[2], i3 = idx4[3];
  const float* __restrict__ kbase =
      kseq + (long)(b * HEADS + head) * NREG * RSZ * HEAD_DIM;
  const float* __restrict__ vbase =
      vseq + (long)(b * HEADS + head) * NREG * RSZ * HEAD_DIM;

  // per-lane LDS byte address of the stage buffers (low 32 bits of the
  // generic pointer are the LDS byte offset)
  const unsigned sTaddr = (unsigned)(uintptr_t)(void*)&sT[0][0] + lane * 16u;

  __syncthreads();

  // ---- Phase 1: S[16,256] = Q Kg^T, K tiles streamed async ----
  const int mb = (lane >> 4) << 3;
  {
    // prime tile 0
    {
      const int n0 = 0, regn = i0;
      async_copy_4k(sTaddr, kbase + ((long)regn * RSZ + (n0 & 63)) * HEAD_DIM + lane * 4);
    }
    for (int ntile = 0; ntile < 16; ++ntile) {
      const int buf = ntile & 1;
      if (ntile < 15) {
        const int n0   = (ntile + 1) * 16;
        const int regn = (n0 < 64) ? i0 : (n0 < 128) ? i1 : (n0 < 192) ? i2 : i3;
        async_copy_4k(sTaddr + (unsigned)((buf ^ 1) * 4096),
                      kbase + ((long)regn * RSZ + (n0 & 63)) * HEAD_DIM + lane * 4);
        wait_async_le8();                       // current tile's 8 done
      } else {
        wait_async_0();
      }
      const float* __restrict__ kt = &sT[buf][0];  // tile row = token n-lrow
      v8f acc = {};
#pragma unroll
      for (int k0 = 0; k0 < HEAD_DIM; k0 += 4) {
        v2f a; a.x = sQ[lrow * QSTR + k0 + kb2]; a.y = sQ[lrow * QSTR + k0 + kb2 + 1];
        v2f bf; bf.x = kt[lrow * 64 + k0 + kb2]; bf.y = kt[lrow * 64 + k0 + kb2 + 1];
        acc = wmma_f32x4(a, bf, acc);
      }
      const int n = ntile * 16 + lrow;
#pragma unroll
      for (int r = 0; r < 8; ++r) sS[(mb + r) * SSTR + n] = acc[r];
    }
  }
  __syncthreads();

  // ---- Phase 2: softmax over 256 keys; lane = (row, half) ----
  {
    float* __restrict__ srow = sS + lrow * SSTR + (lane >> 4) * 128;
    float mx = -INFINITY;
    for (int c = 0; c < 128; ++c) mx = fmaxf(mx, srow[c]);
    mx = fmaxf(mx, __shfl_xor(mx, 16, 32));
    float sum = 0.f;
    for (int c = 0; c < 128; ++c) { float e = __expf(srow[c] - mx); srow[c] = e; sum += e; }
    sum += __shfl_xor(sum, 16, 32);
    float inv = 1.f / sum;
    for (int c = 0; c < 128; ++c) srow[c] *= inv;
  }
  __syncthreads();

  // ---- Phase 3: O[16,64] = P Vg, V tiles streamed async, k-tile outermost ----
  v8f acc[4] = {v8f{}, v8f{}, v8f{}, v8f{}};
  {
    // prime tile 0
    async_copy_4k(sTaddr, vbase + (long)i0 * RSZ * HEAD_DIM + lane * 4);
    for (int kt = 0; kt < 16; ++kt) {
      const int buf = kt & 1;
      if (kt < 15) {
        const int n0   = (kt + 1) * 16;
        const int regn = (n0 < 64) ? i0 : (n0 < 128) ? i1 : (n0 < 192) ? i2 : i3;
        async_copy_4k(sTaddr + (unsigned)((buf ^ 1) * 4096),
                      vbase + ((long)regn * RSZ + (n0 & 63)) * HEAD_DIM + lane * 4);
        wait_async_le8();
      } else {
        wait_async_0();
      }
      const float* __restrict__ vt = &sT[buf][0];  // row = local k, col = d
#pragma unroll
      for (int k0 = 0; k0 < 16; k0 += 4) {
        const int kk = k0 + kb2;                   // even; kk+1 stays in tile
        v2f a; a.x = sS[lrow * SSTR + kt * 16 + kk];
               a.y = sS[lrow * SSTR + kt * 16 + kk + 1];
#pragma unroll
        for (int nt2 = 0; nt2 < 4; ++nt2) {
          const int col = nt2 * 16 + lrow;
          v2f bf; bf.x = vt[kk * 64 + col]; bf.y = vt[(kk + 1) * 64 + col];
          acc[nt2] = wmma_f32x4(a, bf, acc[nt2]);
        }
      }
    }
  }

  // ---- store O (pixel-major into attnout[25088,512]) ----
  const int rh_i = reg / 7, rw_i = reg % 7;
#pragma unroll
  for (int nt2 = 0; nt2 < 4; ++nt2) {
    const int col = nt2 * 16 + lrow;
#pragma unroll
    for (int r = 0; r < 8; ++r) {
      int m  = mblk * 16 + mb + r;               // token in region
      int hh = rh_i * 8 + (m >> 3), ww = rw_i * 8 + (m & 7);
      long rowg = (long)b * NPIX + hh * WW + ww;
      attnout[rowg * DIM + head * HEAD_DIM + col] = acc[nt2][r];
    }
  }
}

// ---------------------------------------------------------------------------
// Kernel 5: LEPE depthwise 3x3 on v (seq layout) + bias, added into attnout.
// ---------------------------------------------------------------------------
__global__ void lepe_kernel(const float* __restrict__ vseq,
                            const float* __restrict__ lw,
                            const float* __restrict__ lb,
                            float* __restrict__ attnout) {
  long i = (long)blockIdx.x * blockDim.x + threadIdx.x;  // [B,NPIX,DIM]
  if (i >= (long)NB * NPIX * DIM) return;
  int c   = (int)(i % DIM);
  int pix = (int)((i / DIM) % NPIX);
  int b   = (int)(i / ((long)DIM * NPIX));
  int hh = pix / WW, ww = pix % WW;
  int head = c >> 6, d = c & 63;
  const float* __restrict__ vb =
      vseq + (long)(b * HEADS + head) * NREG * RSZ * HEAD_DIM + d;
  const float* __restrict__ wt = lw + (long)c * 9;
  float s = lb[c];
#pragma unroll
  for (int ky = 0; ky < 3; ++ky) {
#pragma unroll
    for (int kx = 0; kx < 3; ++kx) {
      int y = hh + ky - 1, x = ww + kx - 1;
      if (y >= 0 && y < HH && x >= 0 && x < WW) {
        int reg = (y >> 3) * 7 + (x >> 3);
        int tok = ((y & 7) << 3) + (x & 7);
        s += vb[((long)reg * RSZ + tok) * HEAD_DIM] * wt[ky * 3 + kx];
      }
    }
  }
  attnout[i] += s;
}

// ---------------------------------------------------------------------------
// Kernel 6: output projection  out[25088,512] = attnout * out_w^T + out_b
// (row order is already NHWC pixel-major, so this writes d_out directly).
// ---------------------------------------------------------------------------
__global__ __launch_bounds__(256) void out_gemm_kernel(
    const float* __restrict__ A, const float* __restrict__ W,
    const float* __restrict__ bias, float* __restrict__ out) {
  const int lane = threadIdx.x & 31;
  const int wid  = blockIdx.x * 8 + (threadIdx.x >> 5);
  const int NT   = DIM / 16;                 // 32
  const int mt   = wid / NT;
  const int nt   = wid % NT;
  const int lrow = lane & 15;
  const int kb   = (lane >> 4) << 1;
  const int row  = mt * 16 + lrow;
  const int col  = nt * 16 + lrow;
  const float* __restrict__ arow = A + (long)row * DIM;
  const float* __restrict__ brow = W + (long)col * DIM;

  v8f acc = {};
#pragma unroll 4
  for (int k0 = 0; k0 < DIM; k0 += 4) {
    v2f a; a.x = arow[k0 + kb]; a.y = arow[k0 + kb + 1];
    v2f b; b.x = brow[k0 + kb]; b.y = brow[k0 + kb + 1];
    acc = wmma_f32x4(a, b, acc);
  }
  const float bn = bias[col];
  const int mbase = mt * 16 + ((lane >> 4) << 3);
#pragma unroll
  for (int r = 0; r < 8; ++r)
    out[(long)(mbase + r) * DIM + col] = acc[r] + bn;
}

// ---------------------------------------------------------------------------
extern "C" void kernel_launch(void* const* d_in, const int* in_sizes, int n_in,
                              void* d_out, int out_size, void* d_ws, size_t ws_size,
                              hipStream_t stream) {
  const float* x      = (const float*)d_in[0];
  const float* qkv_w  = (const float*)d_in[1];
  const float* qkv_b  = (const float*)d_in[2];
  const float* lepe_w = (const float*)d_in[3];
  const float* lepe_b = (const float*)d_in[4];
  const float* out_w  = (const float*)d_in[5];
  const float* out_b  = (const float*)d_in[6];
  float* out = (float*)d_out;

  const long SEQ = (long)NB * HEADS * NREG * RSZ * HEAD_DIM;  // 12,845,056 floats
  float* ws      = (float*)d_ws;
  float* qseq    = ws;
  float* kseq    = ws + SEQ;
  float* vseq    = ws + 2 * SEQ;
  float* attnout = ws + 3 * SEQ;                              // [25088,512]
  float* qr      = ws + 4 * SEQ;                              // [8,49,512]
  float* kr      = qr + (long)NB * NREG * DIM;
  int*   idx     = (int*)(kr + (long)NB * NREG * DIM);        // [8,49,4]

  // 1) QKV projection: 1568*96 = 150528 wave-tiles, 8 waves/block
  qkv_gemm_kernel<<<150528 / 8, 256, 0, stream>>>(x, qkv_w, qkv_b, qseq, kseq, vseq);

  // 2) region pooling
  pool_kernel<<<(NB * NREG * DIM + 255) / 256, 256, 0, stream>>>(qseq, kseq, qr, kr);

  // 3) routing top-k
  route_kernel<<<NB * NREG, 64, 0, stream>>>(qr, kr, idx);

  // 4) gathered attention: B*HEADS*NREG*4 single-wave blocks
  attn_kernel<<<NB * HEADS * NREG * 4, 32, 0, stream>>>(qseq, kseq, vseq, idx, attnout);

  // 5) LEPE depthwise conv added into attnout
  lepe_kernel<<<(int)(((long)NB * NPIX * DIM + 255) / 256), 256, 0, stream>>>(
      vseq, lepe_w, lepe_b, attnout);

  // 6) output projection -> d_out (NHWC)
  out_gemm_kernel<<<50176 / 8, 256, 0, stream>>>(attnout, out_w, out_b, out);
}